// net_89163521065252
// MI455X (gfx1250) — compile-verified
//
#include <hip/hip_runtime.h>
#include <stdint.h>

// ---------------- static problem configuration ----------------
#define N_ANCHORS   8388608
#define ANCH_PER_BLK 2048          // 256 threads * 8 anchors
#define NPOS_MAX    128            // actual positives = 100
#define POOL        2000           // N_POS * SHEM_POOLSIZE
#define NSEL        100
#define NBINS       4096
#define CAND_CAP    4096
#define D_MIN_F     (-16.0f)
#define D_SCALE_F   (128.0f)       // NBINS / 32

// ---------------- workspace layout (bytes) ----------------
#define WS_HIST     0              // uint32[4096]            (16384 B)
#define WS_CANDCNT  16384          // uint32
#define WS_POSCNT   16388          // uint32
#define WS_THRBIN   16392          // int
#define WS_POSREC   16400          // PosRec[128] * 16 B
#define WS_CAND     20480          // Cand[4096]  * 8 B
#define WS_ZERO_BYTES 16400

struct PosRec { int idx; float l0; float l1; int pad; };
struct Cand   { float d; int idx; };

__device__ __forceinline__ float softplus_f(float x) {
    // stable log(1+exp(x))
    float ax = fabsf(x);
    return fmaxf(x, 0.0f) + log1pf(expf(-ax));
}

// Issue one 16B-per-lane async global->LDS copy (CDNA5 GV-mode DMA).
__device__ __forceinline__ void async_copy_b128(uint32_t ldsoff, const char* src) {
    asm volatile("global_load_async_to_lds_b128 %0, %1, off"
                 :: "v"(ldsoff), "v"(src) : "memory");
}
__device__ __forceinline__ void async_wait0() {
    asm volatile("s_wait_asynccnt 0x0" ::: "memory");
}

// =====================================================================
// Kernel 1: streaming scan (HBM-bound pass). Async-stage the block's
// 16 KB logits tile to LDS, overlap with the matches stream, histogram
// d = l1 - l0 for negatives, append positive records.
// =====================================================================
__global__ __launch_bounds__(256) void k_scan(
    const int* __restrict__ matches, const float* __restrict__ logits,
    uint32_t* __restrict__ ghist, uint32_t* __restrict__ pos_cnt,
    PosRec* __restrict__ posrec)
{
    __shared__ uint32_t sh_hist[NBINS];              // 16 KB
    __shared__ float    sh_logits[ANCH_PER_BLK * 2]; // 16 KB

    const int t    = threadIdx.x;
    const int base = blockIdx.x * ANCH_PER_BLK;

    for (int b = t; b < NBINS; b += 256) sh_hist[b] = 0;

    // ---- async global->LDS staging of this block's logits tile ----
    {
        const char* gsrc  = (const char*)(logits + (size_t)base * 2);
        uint32_t    lbase = (uint32_t)(uintptr_t)(&sh_logits[0]); // LDS byte offset
#pragma unroll
        for (int c = 0; c < 4; ++c) {
            int chunk = t * 4 + c;                    // 1024 chunks of 16 B
            async_copy_b128(lbase + (uint32_t)chunk * 16u, gsrc + (size_t)chunk * 16);
        }
    }

    // Overlap: regular vectorized loads of my 8 matches while DMA runs.
    const int a0 = base + t * 8;
    int4 m01 = ((const int4*)(matches + a0))[0];
    int4 m23 = ((const int4*)(matches + a0))[1];
    int m[8] = {m01.x, m01.y, m01.z, m01.w, m23.x, m23.y, m23.z, m23.w};

    async_wait0();
    __syncthreads();

#pragma unroll
    for (int k = 0; k < 8; ++k) {
        int mk = m[k];
        if (mk == 0) continue;                      // neutral: excluded
        float l0 = sh_logits[(t * 8 + k) * 2 + 0];
        float l1 = sh_logits[(t * 8 + k) * 2 + 1];
        if (mk == -1) {                             // negative: histogram d
            float d = l1 - l0;
            int bin = (int)((d - D_MIN_F) * D_SCALE_F);
            bin = min(max(bin, 0), NBINS - 1);
            atomicAdd(&sh_hist[bin], 1u);
        } else {                                    // positive: append record
            uint32_t p = atomicAdd(pos_cnt, 1u);
            if (p < NPOS_MAX) {
                posrec[p].idx = a0 + k;
                posrec[p].l0  = l0;
                posrec[p].l1  = l1;
            }
        }
    }
    __syncthreads();
    for (int b = t; b < NBINS; b += 256) {
        uint32_t v = sh_hist[b];
        if (v) atomicAdd(&ghist[b], v);
    }
}

// =====================================================================
// Kernel 2: find threshold bin so that count(d-bin >= thr) >= POOL.
// =====================================================================
__global__ __launch_bounds__(256) void k_thresh(
    const uint32_t* __restrict__ ghist, int* __restrict__ thr_bin)
{
    __shared__ uint32_t h[NBINS];
    for (int b = threadIdx.x; b < NBINS; b += 256) h[b] = ghist[b];
    __syncthreads();
    if (threadIdx.x == 0) {
        uint32_t acc = 0; int thr = 0;
        for (int b = NBINS - 1; b >= 0; --b) {
            acc += h[b];
            if (acc >= (uint32_t)POOL) { thr = b; break; }
        }
        *thr_bin = thr;
    }
}

// =====================================================================
// Kernel 3: second streaming pass (L2-resident after pass 1 — 100 MB
// working set fits the 192 MB L2). Same async LDS staging; collect all
// negatives at/above the threshold bin.
// =====================================================================
__global__ __launch_bounds__(256) void k_collect(
    const int* __restrict__ matches, const float* __restrict__ logits,
    const int* __restrict__ thr_bin_p, uint32_t* __restrict__ cand_cnt,
    Cand* __restrict__ cand)
{
    __shared__ float sh_logits[ANCH_PER_BLK * 2];    // 16 KB

    const int t    = threadIdx.x;
    const int base = blockIdx.x * ANCH_PER_BLK;
    const int a0   = base + t * 8;

    // ---- async global->LDS staging of this block's logits tile ----
    {
        const char* gsrc  = (const char*)(logits + (size_t)base * 2);
        uint32_t    lbase = (uint32_t)(uintptr_t)(&sh_logits[0]);
#pragma unroll
        for (int c = 0; c < 4; ++c) {
            int chunk = t * 4 + c;
            async_copy_b128(lbase + (uint32_t)chunk * 16u, gsrc + (size_t)chunk * 16);
        }
    }

    // Overlap: matches + threshold scalar while DMA runs.
    const int thr = *thr_bin_p;
    int4 m01 = ((const int4*)(matches + a0))[0];
    int4 m23 = ((const int4*)(matches + a0))[1];
    int m[8] = {m01.x, m01.y, m01.z, m01.w, m23.x, m23.y, m23.z, m23.w};

    async_wait0();
    __syncthreads();

#pragma unroll
    for (int k = 0; k < 8; ++k) {
        if (m[k] != -1) continue;
        float l0 = sh_logits[(t * 8 + k) * 2 + 0];
        float l1 = sh_logits[(t * 8 + k) * 2 + 1];
        float d  = l1 - l0;
        int bin = (int)((d - D_MIN_F) * D_SCALE_F);
        bin = min(max(bin, 0), NBINS - 1);
        if (bin >= thr) {
            uint32_t p = atomicAdd(cand_cnt, 1u);
            if (p < CAND_CAP) { cand[p].d = d; cand[p].idx = a0 + k; }
        }
    }
}

// =====================================================================
// Kernel 4 (single block): sort candidates, SHEM-sample, positives,
// bbox smooth-L1, final scalar.
// =====================================================================
__device__ __forceinline__ float block_sum512(float v, float* red) {
    red[threadIdx.x] = v; __syncthreads();
    for (int s = 256; s > 0; s >>= 1) {
        if ((int)threadIdx.x < s) red[threadIdx.x] += red[threadIdx.x + s];
        __syncthreads();
    }
    float r = red[0]; __syncthreads();
    return r;
}

__global__ __launch_bounds__(512) void k_final(
    const int* __restrict__ matches,
    const float* __restrict__ pred_deltas, const float* __restrict__ target_deltas,
    const uint32_t* __restrict__ cand_cnt, const Cand* __restrict__ cand,
    const uint32_t* __restrict__ pos_cnt, const PosRec* __restrict__ posrec,
    float* __restrict__ out)
{
    __shared__ float sd[CAND_CAP];   // 16 KB
    __shared__ int   si[CAND_CAP];   // 16 KB
    __shared__ int   perm[POOL];     // 8 KB
    __shared__ float red[512];       // 2 KB

    const int n = (int)min(*cand_cnt, (uint32_t)CAND_CAP);
    for (int i = threadIdx.x; i < CAND_CAP; i += 512) {
        if (i < n) { sd[i] = cand[i].d; si[i] = cand[i].idx; }
        else       { sd[i] = -INFINITY; si[i] = 0x7fffffff; }
    }
    __syncthreads();

    // Bitonic sort, descending by d, ties broken by ascending anchor index
    for (int k = 2; k <= CAND_CAP; k <<= 1) {
        for (int j = k >> 1; j > 0; j >>= 1) {
            for (int t = threadIdx.x; t < CAND_CAP / 2; t += 512) {
                int i   = 2 * t - (t & (j - 1));
                int ixj = i | j;
                bool descBlock = ((i & k) == 0);
                float di = sd[i], dj = sd[ixj];
                int   ii = si[i], ij = si[ixj];
                bool before = (di > dj) || (di == dj && ii < ij);
                if (descBlock ? !before : before) {
                    sd[i] = dj; sd[ixj] = di; si[i] = ij; si[ixj] = ii;
                }
            }
            __syncthreads();
        }
    }

    // Fixed deterministic permutation of the 2000-entry pool (stand-in for
    // jax.random.permutation(key(42), 2000) — identical on every call).
    if (threadIdx.x == 0) {
        for (int i = 0; i < POOL; ++i) perm[i] = i;
        uint32_t s = 0x9E3779B9u ^ 42u;
        for (int i = POOL - 1; i > 0; --i) {
            s = s * 1664525u + 1013904223u;
            int j = (int)(s % (uint32_t)(i + 1));
            int tmp = perm[i]; perm[i] = perm[j]; perm[j] = tmp;
        }
    }
    __syncthreads();

    // negative loss over 100 sampled pool entries
    float negAcc = 0.0f;
    if (threadIdx.x < NSEL) negAcc = softplus_f(sd[perm[threadIdx.x]]);

    // positive CE + bbox smooth-L1 (rank = position among sorted pos indices)
    const int pc = (int)min(*pos_cnt, (uint32_t)NPOS_MAX);
    float ceAcc = 0.0f, bbAcc = 0.0f;
    if ((int)threadIdx.x < pc) {
        PosRec r = posrec[threadIdx.x];
        int rank = 0;
        for (int u = 0; u < pc; ++u) rank += (posrec[u].idx < r.idx);
        int tc = matches[r.idx];
        float mx  = fmaxf(r.l0, r.l1);
        float lse = mx + logf(expf(r.l0 - mx) + expf(r.l1 - mx));
        float lt  = (tc == 0) ? r.l0 : r.l1;   // C == 2
        ceAcc = lse - lt;
#pragma unroll
        for (int kk = 0; kk < 4; ++kk) {
            float x = pred_deltas[(size_t)r.idx * 4 + kk] - target_deltas[rank * 4 + kk];
            float ax = fabsf(x);
            bbAcc += (ax < 1.0f) ? 0.5f * x * x : ax - 0.5f;
        }
    }

    float negSum = block_sum512(negAcc, red);
    float ceSum  = block_sum512(ceAcc, red);
    float bbSum  = block_sum512(bbAcc, red);

    if (threadIdx.x == 0) {
        float pcf = (float)max(pc, 1);
        float pos_loss  = ceSum / pcf;
        float neg_loss  = negSum / (float)NSEL;
        float bbox_loss = bbSum / (4.0f * pcf);
        out[0] = 0.5f * (pos_loss + neg_loss) + bbox_loss;
    }
}

// =====================================================================
extern "C" void kernel_launch(void* const* d_in, const int* in_sizes, int n_in,
                              void* d_out, int out_size, void* d_ws, size_t ws_size,
                              hipStream_t stream) {
    (void)in_sizes; (void)n_in; (void)out_size; (void)ws_size;
    const int*   matches       = (const int*)d_in[0];
    const float* logits        = (const float*)d_in[1];
    const float* pred_deltas   = (const float*)d_in[2];
    const float* target_deltas = (const float*)d_in[3];

    char* ws = (char*)d_ws;
    uint32_t* ghist    = (uint32_t*)(ws + WS_HIST);
    uint32_t* cand_cnt = (uint32_t*)(ws + WS_CANDCNT);
    uint32_t* pos_cnt  = (uint32_t*)(ws + WS_POSCNT);
    int*      thr_bin  = (int*)(ws + WS_THRBIN);
    PosRec*   posrec   = (PosRec*)(ws + WS_POSREC);
    Cand*     cand     = (Cand*)(ws + WS_CAND);

    hipMemsetAsync(ws, 0, WS_ZERO_BYTES, stream);

    dim3 grid(N_ANCHORS / ANCH_PER_BLK);   // 4096 blocks
    k_scan   <<<grid, 256, 0, stream>>>(matches, logits, ghist, pos_cnt, posrec);
    k_thresh <<<1,    256, 0, stream>>>(ghist, thr_bin);
    k_collect<<<grid, 256, 0, stream>>>(matches, logits, thr_bin, cand_cnt, cand);
    k_final  <<<1,    512, 0, stream>>>(matches, pred_deltas, target_deltas,
                                        cand_cnt, cand, pos_cnt, posrec, (float*)d_out);
}